// ProposalLayer_89103391523050
// MI455X (gfx1250) — compile-verified
//
#include <hip/hip_runtime.h>
#include <hip/hip_bf16.h>
#include <stdint.h>

// ---------------- problem constants (from reference) ----------------
#define B_SIZE   8
#define N_ANCH   262144
#define K_TOP    6000
#define K_PAD    6144          // 6*1024
#define P_CNT    1000
#define NBINS    4096
#define CAND_CAP 8192
#define NEGV     (-1000000000.0f)
#define NMS_T    0.7f

// ---------------- workspace layout (bytes) ----------------
#define OFF_HIST     0u                                   // B*NBINS*4   = 131072
#define OFF_SELCNT   (OFF_HIST    + B_SIZE*NBINS*4u)      // B*4
#define OFF_CANDCNT  (OFF_SELCNT  + B_SIZE*4u)
#define OFF_CUTBIN   (OFF_CANDCNT + B_SIZE*4u)
#define OFF_ABOVE    (OFF_CUTBIN  + B_SIZE*4u)
#define OFF_ZERO_END (OFF_ABOVE   + B_SIZE*4u)            // = 131200, zeroed each call
#define OFF_SELIDX   131200u                              // B*K_PAD*4 = 196608
#define OFF_SELSC    (OFF_SELIDX  + B_SIZE*K_PAD*4u)      // 327808
#define OFF_CANDKEY  (OFF_SELSC   + B_SIZE*K_PAD*4u)      // 524416 (8B aligned)
#define OFF_BOXES    (OFF_CANDKEY + B_SIZE*CAND_CAP*8u)   // 1048704 (16B aligned)
// total ~1.84 MB

// ---------------- CDNA5 async global->LDS copy (ASYNCcnt path) ----------------
// Per-lane 16B copy into LDS; generic pointers to __shared__ carry the LDS
// byte offset in their low 32 bits on AMDGCN.
__device__ __forceinline__ void async_b128(uint32_t lds_byte_off, const void* gaddr) {
  asm volatile("global_load_async_to_lds_b128 %0, %1, off"
               :: "v"(lds_byte_off), "v"(gaddr) : "memory");
}
__device__ __forceinline__ void wait_async0() { asm volatile("s_wait_asynccnt 0" ::: "memory"); }
__device__ __forceinline__ void wait_async1() { asm volatile("s_wait_asynccnt 1" ::: "memory"); }

__device__ __forceinline__ int clamp_bin(float s) {
  int b = (int)(s * (float)NBINS);
  b = b < 0 ? 0 : b;
  return b > (NBINS - 1) ? (NBINS - 1) : b;
}

// sortable-uint mapping for float (monotonic): f0 < f1  <=>  flip(f0) < flip(f1)
__device__ __forceinline__ uint32_t flip32(float f) {
  uint32_t b = __float_as_uint(f);
  return (b & 0x80000000u) ? ~b : (b | 0x80000000u);
}
__device__ __forceinline__ float unflip32(uint32_t u) {
  uint32_t b = (u & 0x80000000u) ? (u ^ 0x80000000u) : ~u;
  return __uint_as_float(b);
}

// ---------------- K1: streaming histogram of scores (async-LDS pipeline) ---
// 64 chunks/batch * 4096 elems; per step each lane async-copies its own 16B
// (two float2 score pairs) into a double-buffered LDS tile. Each wave only
// reads LDS bytes its own lanes wrote, so only s_wait_asynccnt is needed
// (no block barrier inside the pipeline).
__global__ __launch_bounds__(256)
void k1_hist(const float2* __restrict__ probs, uint32_t* __restrict__ hist) {
  __shared__ uint32_t h[NBINS];        // 16 KB
  __shared__ float4   tile[2][256];    //  8 KB double buffer
  const int t = threadIdx.x;
  const int b = blockIdx.y;
  const int chunk = blockIdx.x;        // 4096 elements per chunk

  for (int i = t; i < NBINS; i += 256) h[i] = 0u;
  __syncthreads();

  const char* src = (const char*)(probs + (size_t)b * N_ANCH + (size_t)chunk * 4096);
  const uint32_t l0 = (uint32_t)(uintptr_t)&tile[0][t];
  const uint32_t l1 = (uint32_t)(uintptr_t)&tile[1][t];

  async_b128(l0, src + t * 16);                       // prologue: tile 0
  #pragma unroll
  for (int s = 0; s < 8; ++s) {                       // 8 tiles * 512 elems = 4096
    if (s + 1 < 8) {
      async_b128(((s + 1) & 1) ? l1 : l0, src + (s + 1) * 4096 + t * 16);
      wait_async1();                                  // oldest tile landed (in-order)
    } else {
      wait_async0();
    }
    float4 v = ((s & 1) ? tile[1] : tile[0])[t];      // (p0,p1,p0,p1)
    atomicAdd(&h[clamp_bin(v.y)], 1u);
    atomicAdd(&h[clamp_bin(v.w)], 1u);
  }
  __syncthreads();
  uint32_t* hg = hist + (size_t)b * NBINS;
  for (int i = t; i < NBINS; i += 256)
    if (h[i]) atomicAdd(&hg[i], h[i]);
}

// ---------------- K0: zero histogram + counters ----------------
__global__ void k0_zero(uint32_t* p, int n) {
  int i = blockIdx.x * blockDim.x + threadIdx.x;
  if (i < n) p[i] = 0u;
}

// ---------------- K2: suffix scan of histogram -> cutoff bin ----------------
__global__ __launch_bounds__(256)
void k2_scan(const uint32_t* __restrict__ hist, uint32_t* cutBin, uint32_t* above) {
  __shared__ uint32_t hl[NBINS];
  const int b = blockIdx.x, t = threadIdx.x;
  const uint32_t* hb = hist + (size_t)b * NBINS;
  for (int i = t; i < NBINS; i += 256) hl[i] = hb[i];
  __syncthreads();
  if (t == 0) {
    uint32_t cum = 0, cut = 0, ab = 0;
    for (int j = NBINS - 1; j >= 0; --j) {
      uint32_t c2 = cum + hl[j];
      if (c2 >= (uint32_t)K_TOP) { cut = (uint32_t)j; ab = cum; break; }
      cum = c2;
    }
    cutBin[b] = cut; above[b] = ab;
  }
}

// ---------------- K3: compact winners; stash boundary-bin candidates -------
__global__ __launch_bounds__(256)
void k3_compact(const float2* __restrict__ probs, const uint32_t* __restrict__ cutBin,
                uint32_t* selCnt, uint32_t* candCnt,
                uint32_t* __restrict__ selIdx, float* __restrict__ selSc,
                unsigned long long* __restrict__ candKey) {
  const int b = blockIdx.y;
  const int base = (blockIdx.x * 256 + threadIdx.x) * 4;
  const int cut = (int)cutBin[b];
  const float2* pb = probs + (size_t)b * N_ANCH;
  uint32_t* sIdx = selIdx + (size_t)b * K_PAD;
  float*    sSc  = selSc  + (size_t)b * K_PAD;
  unsigned long long* cK = candKey + (size_t)b * CAND_CAP;
  #pragma unroll
  for (int e = 0; e < 4; ++e) {
    int i = base + e;
    float s = pb[i].y;
    int bin = clamp_bin(s);
    if (bin > cut) {
      uint32_t pos = atomicAdd(&selCnt[b], 1u);
      if (pos < K_PAD) { sIdx[pos] = (uint32_t)i; sSc[pos] = s; }
    } else if (bin == cut) {
      uint32_t pos = atomicAdd(&candCnt[b], 1u);
      if (pos < CAND_CAP)
        cK[pos] = ((unsigned long long)__float_as_uint(s) << 32) | (uint32_t)(~(uint32_t)i);
    }
  }
}

// ---------------- K4: exact boundary selection via 64-bit radix search -----
// key = (score_bits<<32)|~idx  (score desc, idx asc; all keys distinct).
// Find largest T with count(key>=T) >= need; then select keys >= T.
__global__ __launch_bounds__(256)
void k4_select(const uint32_t* __restrict__ above, uint32_t* selCnt,
               const uint32_t* __restrict__ candCnt,
               const unsigned long long* __restrict__ candKey,
               uint32_t* __restrict__ selIdx, float* __restrict__ selSc) {
  __shared__ uint32_t red[256];
  const int b = blockIdx.x, t = threadIdx.x;
  const unsigned long long* cK = candKey + (size_t)b * CAND_CAP;
  uint32_t cc = candCnt[b];
  const int C = (int)(cc < (uint32_t)CAND_CAP ? cc : (uint32_t)CAND_CAP);
  int need = K_TOP - (int)above[b];
  if (need > C) need = C;
  if (need <= 0) return;                       // uniform across block

  unsigned long long T = 0ull;
  for (int bit = 63; bit >= 0; --bit) {
    const unsigned long long cand = T | (1ull << bit);
    uint32_t c = 0;
    for (int i = t; i < C; i += 256) c += (cK[i] >= cand) ? 1u : 0u;
    red[t] = c; __syncthreads();
    for (int off = 128; off > 0; off >>= 1) {
      if (t < off) red[t] += red[t + off];
      __syncthreads();
    }
    uint32_t tot = red[0]; __syncthreads();
    if (tot >= (uint32_t)need) T = cand;       // uniform decision
  }
  uint32_t* sIdx = selIdx + (size_t)b * K_PAD;
  float*    sSc  = selSc  + (size_t)b * K_PAD;
  for (int i = t; i < C; i += 256) {
    unsigned long long k = cK[i];
    if (k >= T) {
      uint32_t pos = atomicAdd(&selCnt[b], 1u);
      if (pos < K_PAD) {
        sIdx[pos] = ~((uint32_t)k);
        sSc[pos]  = __uint_as_float((uint32_t)(k >> 32));
      }
    }
  }
}

// ---------------- K5: gather deltas/anchors, apply, clip ----------------
__global__ __launch_bounds__(256)
void k5_gather(const float4* __restrict__ bbox, const float4* __restrict__ anch,
               const uint32_t* __restrict__ selCnt, const uint32_t* __restrict__ selIdx,
               float* __restrict__ selSc, float4* __restrict__ boxes) {
  const int b = blockIdx.y;
  const int j = blockIdx.x * 256 + threadIdx.x;
  if (j >= K_TOP) return;
  uint32_t cnt = selCnt[b]; cnt = cnt < (uint32_t)K_TOP ? cnt : (uint32_t)K_TOP;
  float4* ob = boxes + (size_t)b * K_PAD;
  if (j < (int)cnt) {
    uint32_t i = selIdx[(size_t)b * K_PAD + j];
    float4 d = bbox[(size_t)b * N_ANCH + i];
    float4 a = anch[(size_t)b * N_ANCH + i];
    d.x *= 0.1f; d.y *= 0.1f; d.z *= 0.2f; d.w *= 0.2f;      // BBOX_STD
    float h = a.z - a.x, w = a.w - a.y;
    float cy = a.x + 0.5f * h + d.x * h;
    float cx = a.y + 0.5f * w + d.y * w;
    h *= expf(d.z); w *= expf(d.w);
    float y1 = fminf(fmaxf(cy - 0.5f * h, 0.f), 1.f);
    float x1 = fminf(fmaxf(cx - 0.5f * w, 0.f), 1.f);
    float y2 = fminf(fmaxf(cy + 0.5f * h, 0.f), 1.f);
    float x2 = fminf(fmaxf(cx + 0.5f * w, 0.f), 1.f);
    ob[j] = make_float4(y1, x1, y2, x2);
  } else {
    ob[j] = make_float4(0.f, 0.f, 0.f, 0.f);
    selSc[(size_t)b * K_PAD + j] = NEGV;                     // pad slots never win
  }
}

// ---------------- K6: register-resident NMS (1 workgroup / batch) ----------
// 1024 threads x 6 boxes each. Each entry is a sortable u64 key
// (flip(score)<<32 | ~orig_idx) so argmax+tie-break is a single u64 max:
// wave32 shuffle reduce (5 steps) + 32-entry LDS reduce; the winner thread
// identifies itself by key equality (keys unique). 4 barriers / iteration.
__device__ __forceinline__ unsigned long long u64max(unsigned long long a,
                                                     unsigned long long b) {
  return a > b ? a : b;
}

__global__ __launch_bounds__(1024)
void k6_nms(const float4* __restrict__ boxes, const float* __restrict__ selSc,
            const uint32_t* __restrict__ selIdx,
            float* __restrict__ outP, float* __restrict__ outS) {
  __shared__ unsigned long long wK[32];
  __shared__ unsigned long long fK;
  __shared__ float4 bcast;

  const int t = threadIdx.x, b = blockIdx.x;
  const int lane = t & 31, wave = t >> 5;
  const float4*   bb = boxes  + (size_t)b * K_PAD;
  const float*    ss = selSc  + (size_t)b * K_PAD;
  const uint32_t* si = selIdx + (size_t)b * K_PAD;

  const unsigned long long NEGHI = ((unsigned long long)flip32(NEGV)) << 32;

  float4 bx[6]; unsigned long long key[6];
  #pragma unroll
  for (int k = 0; k < 6; ++k) {
    int j = t + k * 1024;
    if (j < K_TOP) {
      bx[k]  = bb[j];
      key[k] = (((unsigned long long)flip32(ss[j])) << 32) | (uint32_t)(~si[j]);
    } else {
      bx[k]  = make_float4(0.f, 0.f, 0.f, 0.f);
      key[k] = NEGHI;                       // ~0xFFFFFFFF == 0 low bits
    }
  }

  float* oP = outP + (size_t)b * P_CNT * 4;
  float* oS = outS + (size_t)b * P_CNT;

  for (int p = 0; p < P_CNT; ++p) {
    // local + wave32 max
    unsigned long long m = key[0];
    #pragma unroll
    for (int k = 1; k < 6; ++k) m = u64max(m, key[k]);
    #pragma unroll
    for (int s = 16; s > 0; s >>= 1)
      m = u64max(m, (unsigned long long)__shfl_xor((long long)m, s, 32));
    if (lane == 0) wK[wave] = m;
    __syncthreads();
    if (t < 32) {
      unsigned long long m2 = wK[t];
      #pragma unroll
      for (int s = 16; s > 0; s >>= 1)
        m2 = u64max(m2, (unsigned long long)__shfl_xor((long long)m2, s, 32));
      if (t == 0) fK = m2;
    }
    __syncthreads();
    const unsigned long long wk = fK;
    const float winS = unflip32((uint32_t)(wk >> 32));
    const bool  valid = winS > 0.5f * NEGV;
    // winner self-identifies (keys unique), publishes its box, self-suppresses
    #pragma unroll
    for (int k = 0; k < 6; ++k) {
      if (key[k] == wk) {
        bcast  = bx[k];
        key[k] = NEGHI | (uint32_t)key[k];   // == idx==i mask in reference
      }
    }
    __syncthreads();
    const float4 wb = bcast;
    if (t == 0) {
      float4 o = valid ? wb : make_float4(0.f, 0.f, 0.f, 0.f);
      oP[p*4+0] = o.x; oP[p*4+1] = o.y; oP[p*4+2] = o.z; oP[p*4+3] = o.w;
      oS[p] = valid ? winS : 0.0f;
    }
    if (valid) {
      const float aw = (wb.z - wb.x) * (wb.w - wb.y);
      #pragma unroll
      for (int k = 0; k < 6; ++k) {
        float y1 = fmaxf(wb.x, bx[k].x);
        float x1 = fmaxf(wb.y, bx[k].y);
        float y2 = fminf(wb.z, bx[k].z);
        float x2 = fminf(wb.w, bx[k].w);
        float inter = fmaxf(y2 - y1, 0.f) * fmaxf(x2 - x1, 0.f);
        float ab = (bx[k].z - bx[k].x) * (bx[k].w - bx[k].y);
        float iou = inter / (aw + ab - inter + 1e-10f);
        if (iou > NMS_T) key[k] = NEGHI | (uint32_t)key[k];
      }
    }
    __syncthreads();
  }
}

// ---------------- launch ----------------
extern "C" void kernel_launch(void* const* d_in, const int* in_sizes, int n_in,
                              void* d_out, int out_size, void* d_ws, size_t ws_size,
                              hipStream_t stream) {
  const float2* probs = (const float2*)d_in[0];   // (B,N,2) f32
  const float4* bbox  = (const float4*)d_in[1];   // (B,N,4) f32
  const float4* anch  = (const float4*)d_in[2];   // (B,N,4) f32

  char* ws = (char*)d_ws;
  uint32_t* hist    = (uint32_t*)(ws + OFF_HIST);
  uint32_t* selCnt  = (uint32_t*)(ws + OFF_SELCNT);
  uint32_t* candCnt = (uint32_t*)(ws + OFF_CANDCNT);
  uint32_t* cutBin  = (uint32_t*)(ws + OFF_CUTBIN);
  uint32_t* above   = (uint32_t*)(ws + OFF_ABOVE);
  uint32_t* selIdx  = (uint32_t*)(ws + OFF_SELIDX);
  float*    selSc   = (float*)(ws + OFF_SELSC);
  unsigned long long* candKey = (unsigned long long*)(ws + OFF_CANDKEY);
  float4*   boxes   = (float4*)(ws + OFF_BOXES);

  float* outP = (float*)d_out;                     // (B,1000,4)
  float* outS = outP + (size_t)B_SIZE * P_CNT * 4; // (B,1000)

  const int nzero = (int)(OFF_ZERO_END / 4);
  k0_zero   <<<dim3((nzero + 255) / 256), dim3(256), 0, stream>>>((uint32_t*)ws, nzero);
  k1_hist   <<<dim3(64, B_SIZE),  dim3(256),  0, stream>>>(probs, hist);
  k2_scan   <<<dim3(B_SIZE),      dim3(256),  0, stream>>>(hist, cutBin, above);
  k3_compact<<<dim3(256, B_SIZE), dim3(256),  0, stream>>>(probs, cutBin, selCnt, candCnt,
                                                           selIdx, selSc, candKey);
  k4_select <<<dim3(B_SIZE),      dim3(256),  0, stream>>>(above, selCnt, candCnt,
                                                           candKey, selIdx, selSc);
  k5_gather <<<dim3((K_TOP + 255) / 256, B_SIZE), dim3(256), 0, stream>>>(bbox, anch, selCnt,
                                                                          selIdx, selSc, boxes);
  k6_nms    <<<dim3(B_SIZE),      dim3(1024), 0, stream>>>(boxes, selSc, selIdx, outP, outS);
}